// Model_53257594470966
// MI455X (gfx1250) — compile-verified
//
#include <hip/hip_runtime.h>

#define NN 4
#define HH 24
#define WW 24
#define TT 4
#define VV 32
#define HWSZ 576                      // H*W
#define NHD 4
#define XSZ (NN*HWSZ*TT*VV)           // 294912 floats per tensor
#define WHSZ (NHD*NN*TT*VV*HWSZ)      // 1179648 floats

typedef __attribute__((ext_vector_type(2))) float v2f;
typedef __attribute__((ext_vector_type(8))) float v8f;
typedef int v4i __attribute__((vector_size(16)));

#if defined(__has_builtin)
#  if __has_builtin(__builtin_amdgcn_global_load_async_to_lds_b128) && \
      __has_builtin(__builtin_amdgcn_s_wait_asynccnt)
#    define USE_ASYNC_LDS 1
#  endif
#endif

__device__ __forceinline__ v8f wmma_k4(v2f a, v2f b, v8f c) {
  // D(16x16,f32) = A(16x4,f32) * B(4x16,f32) + C
  return __builtin_amdgcn_wmma_f32_16x16x4_f32(false, a, false, b, (short)0, c,
                                               false, false);
}

// ---------------------------------------------------------------------------
// Kernel A: Wh[hd][n][s][v][m] = sum_t x[n,h,w,t,v]*mw[t,s] + mb[s]
// grid (72, N, NHD), block 256: tid -> v = tid&31 (coalesced x loads),
// mloc = tid>>5, m = chunk*8+mloc.
// ---------------------------------------------------------------------------
__global__ __launch_bounds__(256) void kA(const float* __restrict__ X,
                                          const float* __restrict__ mapw,
                                          const float* __restrict__ mapb,
                                          float* __restrict__ Wh, int l) {
  const int chunk = blockIdx.x, n = blockIdx.y, hd = blockIdx.z;
  const int tid = threadIdx.x;
  const int v = tid & 31, mloc = tid >> 5;
  const int m = chunk * 8 + mloc;
  const int h = m / 24, w = m - h * 24;
  const float* mw = mapw + (l * NHD + hd) * 16;  // [t][s]
  const float* mb = mapb + (l * NHD + hd) * 4;

  float xv[4];
#pragma unroll
  for (int t0 = 0; t0 < 4; ++t0)
    xv[t0] = X[(((n * HH + h) * WW + w) * TT + t0) * VV + v];

  float* Wp = Wh + (hd * NN + n) * (TT * VV * HWSZ);
#pragma unroll
  for (int s = 0; s < 4; ++s) {
    float acc = mb[s];
#pragma unroll
    for (int t0 = 0; t0 < 4; ++t0) acc = fmaf(xv[t0], mw[t0 * 4 + s], acc);
    Wp[(s * VV + v) * HWSZ + m] = acc;
  }
}

// ---------------------------------------------------------------------------
// Kernel A2: Si[v,t] = sum_{h in [6t,6t+6), w<24, s} Wh[s][v][h*24+w] *
//                      a[(h%6)*192 + w*4 + s]      (Sj uses (w+24)*4+s)
// grid (N, NHD), block 128: v = tid&31, t = tid>>5 (wave-uniform -> s_loads)
// ---------------------------------------------------------------------------
__global__ __launch_bounds__(128) void kA2(const float* __restrict__ Wh,
                                           const float* __restrict__ atv,
                                           float* __restrict__ SiSj, int l) {
  const int n = blockIdx.x, hd = blockIdx.y;
  const int tid = threadIdx.x;
  const int v = tid & 31, t = tid >> 5;
  const float* a = atv + (l * NHD + hd) * 1152;
  const float* Wp = Wh + (hd * NN + n) * (TT * VV * HWSZ);

  float si = 0.0f, sj = 0.0f;
  for (int hh = 0; hh < 6; ++hh) {
    const int h = t * 6 + hh;
    for (int w = 0; w < 24; ++w) {
      const int m = h * 24 + w;
#pragma unroll
      for (int s = 0; s < 4; ++s) {
        const float wv = Wp[(s * VV + v) * HWSZ + m];
        si = fmaf(wv, a[hh * 192 + w * 4 + s], si);
        sj = fmaf(wv, a[hh * 192 + (w + 24) * 4 + s], sj);
      }
    }
  }
  float* S = SiSj + (hd * NN + n) * 256;
  S[v * 4 + t] = si;
  S[128 + v * 4 + t] = sj;
}

// ---------------------------------------------------------------------------
// Kernel B: per (t, n, hd):
//   stage Wh[t] (32x576 f32, 72KB) into LDS via async global->LDS b128
//   adjn (32x32) from B_param;  att[i][j] = softmax_t(lrelu(Si+Sj))[t]
//   M[u][j] = sum_i adjnT[u][i]*att[i][j]          (WMMA, waves 0..3)
//   hp[u][m] = sum_j M[u][j]*WhL[j][m]; elu        (WMMA, 16 waves x 36 mtiles)
// ---------------------------------------------------------------------------
__global__ __launch_bounds__(512) void kB(const float* __restrict__ Wh,
                                          const float* __restrict__ SiSj,
                                          const float* __restrict__ Bparam,
                                          float* __restrict__ hp, int l) {
  __shared__ float WhL[VV * HWSZ];      // 73728 B staged B-matrix tile
  __shared__ float adjT[1024];          // [u][i] = adjn[i][u]
  __shared__ float attL[1024];          // [i][j] (this t)
  __shared__ float Ml[1024];            // [u][j]
  __shared__ float sSi[128], sSj[128];  // [v][t]
  __shared__ float d12s[32], rmn[32], rmx[32];
  __shared__ float sMin, sMax;

  const int t = blockIdx.x, n = blockIdx.y, hd = blockIdx.z;
  const int tid = threadIdx.x;
  const float* Bp = Bparam + (l * NHD + hd) * 1024;
  const float* Wp = Wh + ((hd * NN + n) * TT + t) * (VV * HWSZ);

  // ---- stage the 32x576 Wh tile into LDS (overlapped with att/M work) ----
#ifdef USE_ASYNC_LDS
  for (int idx = tid; idx < (VV * HWSZ) / 4; idx += 512)
    __builtin_amdgcn_global_load_async_to_lds_b128(
        (__attribute__((address_space(1))) v4i*)(Wp + idx * 4),
        (__attribute__((address_space(3))) v4i*)(WhL + idx * 4), 0, 0);
#else
  for (int idx = tid; idx < (VV * HWSZ) / 4; idx += 512) {
    float4 q = ((const float4*)Wp)[idx];
    ((float4*)WhL)[idx] = q;
  }
#endif

  if (tid < 32) {  // per-row min/max of (B + I)
    float mn = 1e30f, mx = -1e30f;
    for (int u = 0; u < 32; ++u) {
      const float val = Bp[tid * 32 + u] + (tid == u ? 1.0f : 0.0f);
      mn = fminf(mn, val);
      mx = fmaxf(mx, val);
    }
    rmn[tid] = mn;
    rmx[tid] = mx;
  }
  if (tid < 128) {
    sSi[tid] = SiSj[(hd * NN + n) * 256 + tid];
    sSj[tid] = SiSj[(hd * NN + n) * 256 + 128 + tid];
  }
  __syncthreads();
  if (tid == 0) {
    float mn = rmn[0], mx = rmx[0];
    for (int i = 1; i < 32; ++i) {
      mn = fminf(mn, rmn[i]);
      mx = fmaxf(mx, rmx[i]);
    }
    sMin = mn;
    sMax = mx;
  }
  __syncthreads();
  const float inv = 1.0f / (sMax - sMin);
  if (tid < 32) {  // D^-1/2 of min-max-normalized adj
    float rs = 0.0f;
    for (int u = 0; u < 32; ++u)
      rs += (Bp[tid * 32 + u] + (tid == u ? 1.0f : 0.0f) - sMin) * inv;
    d12s[tid] = rsqrtf(rs);
  }
  __syncthreads();
  for (int idx = tid; idx < 1024; idx += 512) {
    const int u = idx >> 5, i = idx & 31;
    const float av = (Bp[i * 32 + u] + (i == u ? 1.0f : 0.0f) - sMin) * inv;
    adjT[idx] = d12s[i] * d12s[u] * av;
  }
  for (int idx = tid; idx < 1024; idx += 512) {
    const int i = idx >> 5, j = idx & 31;
    float e[4], mxv = -1e30f;
#pragma unroll
    for (int tt = 0; tt < 4; ++tt) {
      float ev = sSi[i * 4 + tt] + sSj[j * 4 + tt];
      ev = ev > 0.0f ? ev : 0.2f * ev;  // leaky relu
      e[tt] = ev;
      mxv = fmaxf(mxv, ev);
    }
    float sum = 0.0f;
#pragma unroll
    for (int tt = 0; tt < 4; ++tt) {
      e[tt] = __expf(e[tt] - mxv);
      sum += e[tt];
    }
    attL[idx] = e[t] / sum;  // softmax over T, take component t
  }
  __syncthreads();

  const int wv = tid >> 5, lane = tid & 31;
  const int half = lane >> 4, r = lane & 15;

  // --- M = adjT(32x32) @ att(32x32), K = i -------------------------------
  if (wv < 4) {
    const int ut = wv >> 1, jt = wv & 1;
    v8f acc = {};
#pragma unroll
    for (int k = 0; k < 8; ++k) {
      const int i0 = 4 * k + half * 2;
      v2f a, b;
      a.x = adjT[(ut * 16 + r) * 32 + i0];
      a.y = adjT[(ut * 16 + r) * 32 + i0 + 1];
      b.x = attL[i0 * 32 + jt * 16 + r];
      b.y = attL[(i0 + 1) * 32 + jt * 16 + r];
      acc = wmma_k4(a, b, acc);
    }
#pragma unroll
    for (int i = 0; i < 8; ++i)
      Ml[(ut * 16 + i + half * 8) * 32 + jt * 16 + r] = acc[i];
  }
#ifdef USE_ASYNC_LDS
  __builtin_amdgcn_s_wait_asynccnt(0);  // our wave's tile chunks landed
#endif
  __syncthreads();  // everyone's chunks + Ml visible

  // --- hp = Ml(32x32) @ WhL(32x576), K = j; elu; per-head store ----------
  float* Hp = hp + hd * XSZ + n * (HWSZ * TT * VV) + t * VV;
  for (int mt = wv; mt < 36; mt += 16) {
    const int mb = mt * 16;
    v8f c0 = {}, c1 = {};
#pragma unroll
    for (int k = 0; k < 8; ++k) {
      const int j0 = 4 * k + half * 2;
      v2f bF, a0, a1;
      bF.x = WhL[j0 * HWSZ + mb + r];
      bF.y = WhL[(j0 + 1) * HWSZ + mb + r];
      a0.x = Ml[r * 32 + j0];
      a0.y = Ml[r * 32 + j0 + 1];
      a1.x = Ml[(16 + r) * 32 + j0];
      a1.y = Ml[(16 + r) * 32 + j0 + 1];
      c0 = wmma_k4(a0, bF, c0);
      c1 = wmma_k4(a1, bF, c1);
    }
    const int m = mb + r;
#pragma unroll
    for (int i = 0; i < 8; ++i) {
      const int u0 = i + half * 8;
      float v0 = c0[i];
      v0 = v0 > 0.0f ? v0 : __expf(v0) - 1.0f;  // elu
      Hp[m * (TT * VV) + u0] = v0;
      float v1 = c1[i];
      v1 = v1 > 0.0f ? v1 : __expf(v1) - 1.0f;
      Hp[m * (TT * VV) + 16 + u0] = v1;
    }
  }
}

// ---------------------------------------------------------------------------
// Mean over 4 heads (b128 vectorized)
// ---------------------------------------------------------------------------
__global__ __launch_bounds__(256) void kReduce(const float* __restrict__ hp,
                                               float* __restrict__ out) {
  const int idx = blockIdx.x * 256 + threadIdx.x;
  if (idx < XSZ / 4) {
    const float4 a = ((const float4*)hp)[idx];
    const float4 b = ((const float4*)(hp + XSZ))[idx];
    const float4 c = ((const float4*)(hp + 2 * XSZ))[idx];
    const float4 d = ((const float4*)(hp + 3 * XSZ))[idx];
    float4 o;
    o.x = 0.25f * (a.x + b.x + c.x + d.x);
    o.y = 0.25f * (a.y + b.y + c.y + d.y);
    o.z = 0.25f * (a.z + b.z + c.z + d.z);
    o.w = 0.25f * (a.w + b.w + c.w + d.w);
    ((float4*)out)[idx] = o;
  }
}

extern "C" void kernel_launch(void* const* d_in, const int* in_sizes, int n_in,
                              void* d_out, int out_size, void* d_ws,
                              size_t ws_size, hipStream_t stream) {
  const float* x = (const float*)d_in[0];
  const float* map_w = (const float*)d_in[1];
  const float* map_b = (const float*)d_in[2];
  const float* a_temp = (const float*)d_in[3];
  const float* B_param = (const float*)d_in[4];
  float* out = (float*)d_out;

  float* ws = (float*)d_ws;
  float* Wh = ws;                       // WHSZ floats
  float* hpb = ws + WHSZ;               // NHD*XSZ floats (== WHSZ)
  float* SiSj = ws + 2 * (size_t)WHSZ;  // NHD*NN*2*VV*TT = 4096 floats
  float* Xl = SiSj + 4096;              // XSZ floats (layer-1 output)

  for (int l = 0; l < 2; ++l) {
    const float* Xin = (l == 0) ? x : Xl;
    float* Xout = (l == 0) ? Xl : out;
    kA<<<dim3(72, NN, NHD), 256, 0, stream>>>(Xin, map_w, map_b, Wh, l);
    kA2<<<dim3(NN, NHD), 128, 0, stream>>>(Wh, a_temp, SiSj, l);
    kB<<<dim3(TT, NN, NHD), 512, 0, stream>>>(Wh, SiSj, B_param, hpb, l);
    kReduce<<<dim3(XSZ / 4 / 256), 256, 0, stream>>>(hpb, Xout);
  }
}